// FCVI_Net_87909390614699
// MI455X (gfx1250) — compile-verified
//
#include <hip/hip_runtime.h>

// ---------------------------------------------------------------------------
// FCVI net for MI455X (gfx1250, wave32).
//  samples = mean + L @ z  : M=4609, K=4609 (triangular), N=8192, FP32.
//  -> V_WMMA_F32_16X16X4_F32 (exact fp32 matrix pipe).
//  L dense-padded (85.5 MB) lives in L2 (192 MB); z streamed once per WG.
//  MLP epilogue fused per 64-row block -> no 151 MB samples buffer.
// ---------------------------------------------------------------------------

static constexpr int DIN   = 4;
static constexpr int NPAR  = 4609;   // parameter count
static constexpr int NPAD  = 4624;   // NPAR rounded up to multiple of 16
static constexpr int NCOL  = 8192;   // S*B sample columns
static constexpr int BN    = 64;     // columns per workgroup
static constexpr int LSTR  = 68;     // LDS tile row stride (floats): 16B-aligned rows

typedef float v2f __attribute__((ext_vector_type(2)));
typedef float v8f __attribute__((ext_vector_type(8)));

// ---------------------------------------------------------------------------
// Kernel 1: expand cov_vector -> dense padded lower-triangular L with exp(diag)
// ---------------------------------------------------------------------------
__global__ __launch_bounds__(256) void build_L_kernel(const float* __restrict__ cov,
                                                      float* __restrict__ Lp) {
  const int tot4 = (NPAD * NPAD) >> 2;
  const int idx4 = blockIdx.x * 256 + threadIdx.x;
  if (idx4 >= tot4) return;
  const int base = idx4 << 2;
  const int a = base / NPAD;          // row
  const int c = base - a * NPAD;      // col base (multiple of 4)
  float v[4] = {0.f, 0.f, 0.f, 0.f};
  if (a < NPAR) {
    const int ro = (a * (a + 1)) >> 1;  // fits int32 (max ~10.7M)
#pragma unroll
    for (int u = 0; u < 4; ++u) {
      const int b = c + u;
      if (b < a)       v[u] = cov[ro + b];
      else if (b == a) v[u] = __expf(cov[ro + a]);
    }
  }
  float4 f4;
  f4.x = v[0]; f4.y = v[1]; f4.z = v[2]; f4.w = v[3];
  *reinterpret_cast<float4*>(&Lp[(size_t)a * NPAD + c]) = f4;
}

// ---------------------------------------------------------------------------
// Kernel 2: fused  (mean + L @ z)  +  per-sample MLP.
// Grid: 128 WGs x 256 threads (8 wave32). Each WG owns 64 sample columns and
// walks 72 row-blocks of 64 in order; K-tiles only up to the diagonal.
// ---------------------------------------------------------------------------
__global__ __launch_bounds__(256) void fused_sample_mlp_kernel(
    const float* __restrict__ x,     // (256, 4) row-major
    const float* __restrict__ mean,  // (4609)
    const float* __restrict__ z,     // (4609, 8192) row-major
    const float* __restrict__ Lp,    // (NPAD, NPAD) row-major, dense tri
    float* __restrict__ out)         // (8192)
{
  __shared__ float Lt[64 * LSTR];    // L tile  (row-major, stride 68)
  __shared__ float Zt[64 * LSTR];    // z tile transposed: Zt[col][k]
  __shared__ float Sblk[64 * 64];    // finished sample block (rows x cols)
  __shared__ float A0[64 * 64];      // layer-0 activations a0[o][c]
  __shared__ float A1[64 * 64];      // layer-1 activations a1[o][c]
  __shared__ float Red[4 * 64];      // layer-2 partial reduction
  __shared__ float Ms[64];           // mean slice for current row block

  const int t  = threadIdx.x;
  const int ln = t & 15;             // lane % 16
  const int kh = (t >> 4) & 1;       // lane / 16 within wave32
  const int w  = t >> 5;             // wave 0..7
  const int rw = w >> 1;             // 16-row strip within block
  const int cw = w & 1;              // 32-col half within block
  const int c0 = blockIdx.x * BN;    // global column base

  // consumer mapping: 4 threads per column, 16 outputs each
  const int cc = t & 63;
  const int cq = t >> 6;
  const int o0 = cq << 4;

  for (int i = t; i < 64 * 64; i += 256) { A0[i] = 0.f; A1[i] = 0.f; }

  // LDS fragment pointers (constant across the whole kernel)
  const float* ap  = &Lt[(16 * rw + ln) * LSTR + 2 * kh];
  const float* bp0 = &Zt[(32 * cw + ln) * LSTR + 2 * kh];
  const float* bp1 = &Zt[(32 * cw + 16 + ln) * LSTR + 2 * kh];

  for (int m = 0; m <= 71; ++m) {
    v8f acc0 = {0.f, 0.f, 0.f, 0.f, 0.f, 0.f, 0.f, 0.f};
    v8f acc1 = {0.f, 0.f, 0.f, 0.f, 0.f, 0.f, 0.f, 0.f};

    for (int k = 0; k <= m; ++k) {          // triangular: K only to diagonal
      __syncthreads();
      // ---- stage L tile (64x64) and transposed z tile (64 cols x 64 K) ----
#pragma unroll
      for (int j = 0; j < 4; ++j) {
        const int q4 = t + 256 * j;         // float4 index 0..1023
        const int r  = q4 >> 4;             // tile row 0..63
        const int cb = (q4 & 15) << 2;      // tile col 0,4,..,60
        const float4 lv = *reinterpret_cast<const float4*>(
            &Lp[(size_t)(64 * m + r) * NPAD + 64 * k + cb]);
        *reinterpret_cast<float4*>(&Lt[r * LSTR + cb]) = lv;
        const float4 zv = *reinterpret_cast<const float4*>(
            &z[(size_t)(64 * k + r) * NCOL + c0 + cb]);
        Zt[(cb + 0) * LSTR + r] = zv.x;
        Zt[(cb + 1) * LSTR + r] = zv.y;
        Zt[(cb + 2) * LSTR + r] = zv.z;
        Zt[(cb + 3) * LSTR + r] = zv.w;
      }
      __syncthreads();
      if (k < m) {  // warm L2/L0 for next K-tile while we compute
        __builtin_prefetch(&Lp[(size_t)(64 * m + (t & 63)) * NPAD + 64 * (k + 1)], 0, 0);
        __builtin_prefetch(&z[(size_t)(64 * (k + 1) + (t & 63)) * NCOL + c0], 0, 0);
      }
      // ---- 64x64x64 MACs: 16 K-steps x 2 col tiles of v_wmma_f32_16x16x4_f32
#pragma unroll
      for (int kk = 0; kk < 64; kk += 4) {
        const v2f af  = *reinterpret_cast<const v2f*>(ap  + kk);
        const v2f bf0 = *reinterpret_cast<const v2f*>(bp0 + kk);
        const v2f bf1 = *reinterpret_cast<const v2f*>(bp1 + kk);
        acc0 = __builtin_amdgcn_wmma_f32_16x16x4_f32(false, af, false, bf0,
                                                     (short)0, acc0, false, false);
        acc1 = __builtin_amdgcn_wmma_f32_16x16x4_f32(false, af, false, bf1,
                                                     (short)0, acc1, false, false);
      }
    }

    // ---- scatter C fragments to LDS sample block (ISA C layout) ----
#pragma unroll
    for (int j = 0; j < 8; ++j) {
      const int row = 16 * rw + j + 8 * kh;
      Sblk[row * 64 + 32 * cw + ln]      = acc0[j];
      Sblk[row * 64 + 32 * cw + 16 + ln] = acc1[j];
    }
    if (t < 64) Ms[t] = mean[64 * m + t];
    __syncthreads();

    // ---- fused MLP consumption of rows [64m, 64m+64) ----
    if (m <= 3) {                       // w0: rows i*64+o, i == m
      const float xv = x[((c0 + cc) & 255) * DIN + m];
#pragma unroll
      for (int oo = 0; oo < 16; ++oo) {
        const int o = o0 + oo;
        A0[o * 64 + cc] += xv * (Sblk[o * 64 + cc] + Ms[o]);
      }
    } else if (m == 4) {                // b0 + relu
#pragma unroll
      for (int oo = 0; oo < 16; ++oo) {
        const int o = o0 + oo;
        A0[o * 64 + cc] = fmaxf(A0[o * 64 + cc] + Sblk[o * 64 + cc] + Ms[o], 0.f);
      }
    } else if (m <= 68) {               // w1: rows 320 + i*64 + o, i == m-5
      const float av = A0[(m - 5) * 64 + cc];
#pragma unroll
      for (int oo = 0; oo < 16; ++oo) {
        const int o = o0 + oo;
        A1[o * 64 + cc] += av * (Sblk[o * 64 + cc] + Ms[o]);
      }
    } else if (m == 69) {               // b1 + relu
#pragma unroll
      for (int oo = 0; oo < 16; ++oo) {
        const int o = o0 + oo;
        A1[o * 64 + cc] = fmaxf(A1[o * 64 + cc] + Sblk[o * 64 + cc] + Ms[o], 0.f);
      }
    } else if (m == 70) {               // w2: partial dot over i
      float p = 0.f;
#pragma unroll
      for (int oo = 0; oo < 16; ++oo) {
        const int i = o0 + oo;
        p += A1[i * 64 + cc] * (Sblk[i * 64 + cc] + Ms[i]);
      }
      Red[cq * 64 + cc] = p;
    } else {                            // m == 71: b2 (row 4544) + writeout
      if (cq == 0) {
        out[c0 + cc] = Red[cc] + Red[64 + cc] + Red[128 + cc] + Red[192 + cc]
                     + Sblk[cc] + Ms[0];
      }
    }
    __syncthreads();
  }
}

// ---------------------------------------------------------------------------
extern "C" void kernel_launch(void* const* d_in, const int* in_sizes, int n_in,
                              void* d_out, int out_size, void* d_ws, size_t ws_size,
                              hipStream_t stream) {
  (void)in_sizes; (void)n_in; (void)out_size; (void)ws_size;
  const float* x    = (const float*)d_in[0];   // (256, 4)
  const float* cov  = (const float*)d_in[1];   // (N*(N+1)/2)
  const float* mean = (const float*)d_in[2];   // (4609)
  const float* z    = (const float*)d_in[3];   // (4609, 8192)
  float* out = (float*)d_out;                  // (8192)
  float* Lp  = (float*)d_ws;                   // NPAD*NPAD floats = 85.5 MB

  const int tot4 = (NPAD * NPAD) / 4;
  build_L_kernel<<<(tot4 + 255) / 256, 256, 0, stream>>>(cov, Lp);
  fused_sample_mlp_kernel<<<NCOL / BN, 256, 0, stream>>>(x, mean, z, Lp, out);
}